// Decoder_64742337019981
// MI455X (gfx1250) — compile-verified
//
#include <hip/hip_runtime.h>
#include <hip/hip_bf16.h>
#include <math.h>

typedef float v2f __attribute__((ext_vector_type(2)));
typedef float v8f __attribute__((ext_vector_type(8)));

#define HDIM   2048
#define FOURH  8192
#define TILE_M 16
#define KCH    4            // K chunks per row
#define KCHUNK (HDIM / KCH) // 512 columns per chunk
#define NTILE  (FOURH / TILE_M) // 512 row tiles

// ---------------------------------------------------------------------------
// GEMV via V_WMMA_F32_16X16X4_F32.
//   job = (mat, tile, kchunk). Each wave computes a full 512-column partial
//   dot product for 16 rows of W (A fragment) against vector v (B fragment,
//   replicated across all 16 N columns so every D column equals the dot).
//   Accumulation stays inside the WMMA C operand -> zero cross-lane reduces.
//   Results stored to partial[(mat*KCH+kc)*8192 + row] with plain stores
//   (deterministic, no atomics).
// ---------------------------------------------------------------------------
__global__ __launch_bounds__(128) void gemv4h_wmma(
    const float* __restrict__ W0, const float* __restrict__ v0,
    const float* __restrict__ W1, const float* __restrict__ v1,
    float* __restrict__ partial, int jobBase)
{
    const int lane = threadIdx.x & 31;
    const int wave = threadIdx.x >> 5;
    const int job  = jobBase + blockIdx.x * 4 + wave;

    const int kc   = job & (KCH - 1);
    const int tile = (job >> 2) & (NTILE - 1);
    const int mat  = job >> 11;

    const float* __restrict__ W = mat ? W1 : W0;
    const float* __restrict__ v = mat ? v1 : v0;

    const int row0  = tile * TILE_M;
    const int kbase = kc * KCHUNK;
    const int m     = lane & 15;          // row within tile
    const int koff  = (lane >> 4) << 1;   // 0 for lanes 0-15, 2 for 16-31

    // A fragment: lane (m, half) loads W[row0+m][k + koff .. +1]  (8B/lane)
    const float* a = W + (size_t)(row0 + m) * HDIM + kbase + koff;
    // B fragment: v[k + koff .. +1] broadcast across the half-wave (L0 hits)
    const float* b = v + kbase + koff;

    v8f c0 = {0.f, 0.f, 0.f, 0.f, 0.f, 0.f, 0.f, 0.f};
    v8f c1 = {0.f, 0.f, 0.f, 0.f, 0.f, 0.f, 0.f, 0.f};

#pragma unroll 4
    for (int k = 0; k < KCHUNK; k += 8) {
        v2f a0 = *(const v2f*)(a + k);
        v2f b0 = *(const v2f*)(b + k);
        v2f a1 = *(const v2f*)(a + k + 4);
        v2f b1 = *(const v2f*)(b + k + 4);
        // (neg_a, A, neg_b, B, c_mod, C, reuse_a, reuse_b)
        c0 = __builtin_amdgcn_wmma_f32_16x16x4_f32(false, a0, false, b0,
                                                   (short)0, c0, false, false);
        c1 = __builtin_amdgcn_wmma_f32_16x16x4_f32(false, a1, false, b1,
                                                   (short)0, c1, false, false);
    }
    v8f c = c0 + c1;

    // D layout: VGPR r, lanes 0-15 -> M=r ; lanes 16-31 -> M=r+8.
    // B replicated over N, so every lane in a half holds all 8 row results.
    if (m == 0) {
        float* out = partial + (size_t)(mat * KCH + kc) * FOURH + row0
                   + (lane >> 4) * 8;
#pragma unroll
        for (int r = 0; r < 8; ++r) out[r] = c[r];
    }
}

// ---------------------------------------------------------------------------
// Elementwise LSTM cell: sums partials + biases (+ x*Wih0 col for layer 0),
// applies gate math, writes h/c to d_out and h to the next layer's input vec.
// ---------------------------------------------------------------------------
__global__ __launch_bounds__(256) void lstm_elem(
    const float* __restrict__ partial, int pstart,
    const float* __restrict__ wih0col, const float* __restrict__ xscal,
    const float* __restrict__ bih, const float* __restrict__ bhh,
    const float* __restrict__ c_prev,
    float* __restrict__ h_out, float* __restrict__ c_out,
    float* __restrict__ in_next)
{
    const int j = blockIdx.x * blockDim.x + threadIdx.x;
    if (j >= HDIM) return;

    float g[4];
#pragma unroll
    for (int q = 0; q < 4; ++q) {
        const int row = q * HDIM + j;
        float s = bih[row] + bhh[row];
        if (wih0col) s += xscal[0] * wih0col[row];   // layer 0: x @ Wih0.T
        for (int p = pstart; p < 2 * KCH; ++p) s += partial[(size_t)p * FOURH + row];
        g[q] = s;
    }
    const float ig = 1.f / (1.f + expf(-g[0]));
    const float fg = 1.f / (1.f + expf(-g[1]));
    const float gg = tanhf(g[2]);
    const float og = 1.f / (1.f + expf(-g[3]));
    const float c2 = fg * c_prev[j] + ig * gg;
    const float h2 = og * tanhf(c2);

    h_out[j] = h2;
    c_out[j] = c2;
    in_next[j] = h2;
}

// ---------------------------------------------------------------------------
// Output projection: y = Wout . h + bout  (one workgroup, LDS tree reduce)
// ---------------------------------------------------------------------------
__global__ __launch_bounds__(256) void proj_dot(
    const float* __restrict__ Wout, const float* __restrict__ h,
    const float* __restrict__ bout, float* __restrict__ y)
{
    __shared__ float red[256];
    const int t = threadIdx.x;
    float s = 0.f;
    for (int j = t; j < HDIM; j += 256) s += Wout[j] * h[j];
    red[t] = s;
    __syncthreads();
    for (int w = 128; w > 0; w >>= 1) {
        if (t < w) red[t] += red[t + w];
        __syncthreads();
    }
    if (t == 0) y[0] = red[0] + bout[0];
}

// ---------------------------------------------------------------------------
// d_in order: 0:x 1:input_hidden 2:input_cell 3:Wih0 4:Whh0 5:bih0 6:bhh0
//             7:Wih1 8:Whh1 9:bih1 10:bhh1 11:Wih2 12:Whh2 13:bih2 14:bhh2
//             15:Wout 16:bout
// d_out: [ y(1) | hidden (3*2048) | cell (3*2048) ]
// ---------------------------------------------------------------------------
extern "C" void kernel_launch(void* const* d_in, const int* in_sizes, int n_in,
                              void* d_out, int out_size, void* d_ws, size_t ws_size,
                              hipStream_t stream)
{
    const float* x   = (const float*)d_in[0];
    const float* ih  = (const float*)d_in[1];   // [3][2048]
    const float* ic  = (const float*)d_in[2];   // [3][2048]
    const float* Wih0 = (const float*)d_in[3];
    const float* Whh0 = (const float*)d_in[4];
    const float* bih0 = (const float*)d_in[5];
    const float* bhh0 = (const float*)d_in[6];
    const float* Wih1 = (const float*)d_in[7];
    const float* Whh1 = (const float*)d_in[8];
    const float* bih1 = (const float*)d_in[9];
    const float* bhh1 = (const float*)d_in[10];
    const float* Wih2 = (const float*)d_in[11];
    const float* Whh2 = (const float*)d_in[12];
    const float* bih2 = (const float*)d_in[13];
    const float* bhh2 = (const float*)d_in[14];
    const float* Wout = (const float*)d_in[15];
    const float* bout = (const float*)d_in[16];

    float* out   = (float*)d_out;
    float* y     = out;
    float* hid   = out + 1;            // [3][2048]
    float* cel   = out + 1 + 3 * HDIM; // [3][2048]

    float* P     = (float*)d_ws;       // partials: [8][8192] floats (256 KB)
    float* invec = P + 2 * KCH * FOURH;// next-layer input vector (aligned)

    const int elemBlocks = HDIM / 256;

    // ---- layer 0: only Whh0 @ h0 through WMMA (mat=1 jobs only) ----
    gemv4h_wmma<<<NTILE * KCH / 4, 128, 0, stream>>>(
        Whh0, ih, Whh0, ih, P, /*jobBase=*/NTILE * KCH);
    lstm_elem<<<elemBlocks, 256, 0, stream>>>(
        P, /*pstart=*/KCH, Wih0, x, bih0, bhh0, ic,
        hid, cel, invec);

    // ---- layer 1: Wih1 @ h0_out  +  Whh1 @ h1 ----
    gemv4h_wmma<<<2 * NTILE * KCH / 4, 128, 0, stream>>>(
        Wih1, invec, Whh1, ih + HDIM, P, /*jobBase=*/0);
    lstm_elem<<<elemBlocks, 256, 0, stream>>>(
        P, /*pstart=*/0, nullptr, x, bih1, bhh1, ic + HDIM,
        hid + HDIM, cel + HDIM, invec);

    // ---- layer 2: Wih2 @ h1_out  +  Whh2 @ h2 ----
    gemv4h_wmma<<<2 * NTILE * KCH / 4, 128, 0, stream>>>(
        Wih2, invec, Whh2, ih + 2 * HDIM, P, /*jobBase=*/0);
    lstm_elem<<<elemBlocks, 256, 0, stream>>>(
        P, /*pstart=*/0, nullptr, x, bih2, bhh2, ic + 2 * HDIM,
        hid + 2 * HDIM, cel + 2 * HDIM, invec);

    // ---- output projection ----
    proj_dot<<<1, 256, 0, stream>>>(Wout, invec, bout, y);
}